// DTW_Loss_49168785605280
// MI455X (gfx1250) — compile-verified
//
#include <hip/hip_runtime.h>
#include <cstdint>

#ifndef __has_builtin
#define __has_builtin(x) 0
#endif

#if __has_builtin(__builtin_amdgcn_tensor_load_to_lds)
#define HAVE_TDM 1
#else
#define HAVE_TDM 0
#endif

#if defined(__has_include)
#if __has_include(<hip/amd_detail/amd_gfx1250_TDM.h>)
#define TDM_SIX_ARGS 1   // amdgpu-toolchain (clang-23 / therock headers): 6-arg builtin
#else
#define TDM_SIX_ARGS 0   // ROCm 7.2 (clang-22): 5-arg builtin
#endif
#else
#define TDM_SIX_ARGS 0
#endif

constexpr int   B_    = 8;
constexpr int   L_    = 1024;
constexpr int   NW    = 32;           // waves per block (wave32)
constexpr float MAXV  = 100.0f;
// exp(-rho*x) = exp2(x * CEXP), CEXP = -rho*log2(e), rho = 10
constexpr float CEXP  = -14.426950408889634f;
// -(1/rho)*ln(m) = KLOG * log2(m)
constexpr float KLOG  = -0.069314718055994530942f;

#if HAVE_TDM
typedef unsigned int u32x4 __attribute__((ext_vector_type(4)));
typedef int          i32x4 __attribute__((ext_vector_type(4)));
typedef int          i32x8 __attribute__((ext_vector_type(8)));

// DMA one contiguous row of `nelem` floats from global memory into LDS at
// byte offset `lds_byte_off` via the Tensor Data Mover (1D tile, 4B elems).
// D# per CDNA5 ISA ch.8; groups 2/3 zero -> backend emits the 2-group form.
__device__ __forceinline__ void tdm_load_row_to_lds(const float* gsrc,
                                                    uint32_t lds_byte_off,
                                                    uint32_t nelem) {
  const uint64_t ga = (uint64_t)(uintptr_t)gsrc;
  u32x4 g0;
  g0.x = 1u;                                        // count=1, user-mode, no gather
  g0.y = lds_byte_off;                              // lds_addr (bytes)
  g0.z = (uint32_t)(ga & 0xFFFFFFFFu);              // global_addr[31:0]
  g0.w = (uint32_t)((ga >> 32) & 0x01FFFFFFu)       // global_addr[56:32]
         | (2u << 30);                              // type = 2 ("image")
  i32x8 g1;
  g1[0] = (int)(2u << 16);                          // workgroup_mask=0, data_size=2 (4B)
  g1[1] = (int)((nelem & 0xFFFFu) << 16);           // tensor_dim0[15:0]
  g1[2] = (int)(((nelem >> 16) & 0xFFFFu)           // tensor_dim0[31:16]
         | (1u << 16));                             // tensor_dim1 = 1
  g1[3] = (int)((nelem & 0xFFFFu) << 16);           // tile_dim0 = nelem
  g1[4] = 0;                                        // tile_dim1=0 (1D tile), tile_dim2=0
  g1[5] = (int)nelem;                               // tensor_dim0_stride[31:0]
  g1[6] = 0;
  g1[7] = 0;
  i32x4 z4 = {0, 0, 0, 0};
#if TDM_SIX_ARGS
  i32x8 z8 = {0, 0, 0, 0, 0, 0, 0, 0};
  __builtin_amdgcn_tensor_load_to_lds(g0, g1, z4, z4, z8, 0);
#else
  __builtin_amdgcn_tensor_load_to_lds(g0, g1, z4, z4, 0);
#endif
}
#endif  // HAVE_TDM

// Wave-pipelined skewed soft-DTW DP, one workgroup per batch.
//
// Wave w owns padded rows [32w+1, 32w+32]; lane l owns row i = 32w+1+l.
// At wave-local step d (0..L+30), lane l computes padded cell (i, j=d-l+1):
//   left = own result of step d-1 (register prev1)
//   up   = lane (l-1)'s result of step d-1 (shfl_up)   [lane 0: LDS boundary row]
//   diag = lane (l-1)'s result of step d-2 = last step's `up` (register upPrev)
// Lanes hold MAXV outside their valid range, which is exactly the column-0 /
// post-tail border the padded recurrence needs. Lane 31 publishes its row to
// the LDS boundary row consumed by lane 0 of the wave below.
//
// Pipeline: wave w is active in phases p >= 2w, doing 32 steps per phase, one
// __syncthreads per phase. Consumer step d of wave w needs producer step d+31
// of wave w-1: producer phase = 2w-2 + floor((d+31)/32) <= 2w-1 + floor(d/32)
// < 2w + floor(d/32) = consumer phase -> strictly earlier, always safe.
__global__ __launch_bounds__(NW * 32) void softdtw_kernel(const float* __restrict__ out_t,
                                                          const float* __restrict__ tgt_t,
                                                          float* __restrict__ ws) {
  __shared__ float s_tgt[L_];
  __shared__ float bnd[NW - 1][L_ + 1];   // bnd[w] = D[32(w+1)][*], written by wave w lane 31

  const int t    = (int)threadIdx.x;
  const int b    = (int)blockIdx.x;
  const int w    = t >> 5;
  const int lane = t & 31;

  const float* __restrict__ orow = out_t + (size_t)b * L_ * L_;  // output[b,0,:]
  const float* __restrict__ trow = tgt_t + (size_t)b * L_ * L_;  // target[b,0,:]

  // Stage target row into LDS with the Tensor Data Mover.
#if HAVE_TDM
  if (t == 0) {
    const uint32_t lds_off = (uint32_t)(uintptr_t)(void*)&s_tgt[0];  // flat ptr low 32b = LDS offset
    tdm_load_row_to_lds(trow, lds_off, (uint32_t)L_);
    __builtin_amdgcn_s_wait_tensorcnt(0);
  }
#else
  s_tgt[t] = trow[t];
#endif

  const float myout  = orow[t];                 // cost row constant for row i = t+1
  float prev1  = MAXV;                          // D[i][j-1]; column-0 border before start
  float upPrev = (t == 0) ? 0.0f : MAXV;        // diag seed: D[0][0]=0 for wave0 lane0, else MAX
  const float* bndr = (w > 0)      ? bnd[w - 1] : nullptr;
  float*       bndw = (w < NW - 1) ? bnd[w]     : nullptr;
  float last = 0.0f;

  __syncthreads();

  constexpr int totalSteps = L_ + 31;                         // 1055
  constexpr int phases = 2 * (NW - 1) + (totalSteps + 31) / 32;  // 62 + 33 = 95
  for (int p = 0; p < phases; ++p) {
    const int base = (p - 2 * w) * 32;          // wave-uniform
    if (base >= 0 && base < totalSteps) {
      const int lim = (totalSteps - base < 32) ? (totalSteps - base) : 32;
      for (int q = 0; q < lim; ++q) {
        const int d = base + q;
        float up = __shfl_up(prev1, 1, 32);     // lane l-1's step d-1 result
        if (lane == 0) {
          if (w == 0) up = MAXV;                           // D[0][j>=1] = MAX border
          else        up = (d < L_) ? bndr[d + 1] : MAXV;  // D[32w][d+1]
        }
        const float diag = upPrev;              // lane l-1's step d-2 result

        int ci = d - lane;                                 // j-1
        ci = ci < 0 ? 0 : (ci > L_ - 1 ? L_ - 1 : ci);
        const float cost = fabsf(myout - s_tgt[ci]);

        const float m = (__builtin_amdgcn_exp2f(diag  * CEXP) +
                         __builtin_amdgcn_exp2f(prev1 * CEXP) +
                         __builtin_amdgcn_exp2f(up    * CEXP)) * (1.0f / 3.0f);
        const float val = cost + KLOG * __builtin_amdgcn_logf(m + 1e-12f);

        const bool valid = (unsigned)(d - lane) < (unsigned)L_;
        const float nv = valid ? val : MAXV;    // outside range: keep border value

        if (lane == 31 && valid && w < NW - 1)
          bndw[d - 30] = nv;                    // publish bottom row, column j = d-30

        upPrev = up;
        prev1  = nv;
        last   = nv;
      }
    }
    __syncthreads();
  }

  // D[L][L]: wave 31, lane 31, final valid step d = 1054.
  if (t == NW * 32 - 1) ws[b] = last;
}

// Deterministic fixed-order mean over the 8 per-batch corners.
__global__ void reduce_kernel(const float* __restrict__ ws, float* __restrict__ out) {
  if (threadIdx.x == 0 && blockIdx.x == 0) {
    float s = 0.0f;
    for (int i = 0; i < B_; ++i) s += ws[i];
    out[0] = s * (1.0f / (float)B_);
  }
}

extern "C" void kernel_launch(void* const* d_in, const int* in_sizes, int n_in,
                              void* d_out, int out_size, void* d_ws, size_t ws_size,
                              hipStream_t stream) {
  (void)in_sizes; (void)n_in; (void)out_size; (void)ws_size;
  const float* out_t = (const float*)d_in[0];
  const float* tgt_t = (const float*)d_in[1];
  float* ws = (float*)d_ws;

  softdtw_kernel<<<dim3(B_), dim3(NW * 32), 0, stream>>>(out_t, tgt_t, ws);
  reduce_kernel<<<dim3(1), dim3(32), 0, stream>>>(ws, (float*)d_out);
}